// ApproxLinear_80857054314915
// MI455X (gfx1250) — compile-verified
//
#include <hip/hip_runtime.h>
#include <hip/hip_bf16.h>

// ---------------------------------------------------------------------------
// ApproxLinear: scores = x @ W^T  (f32, WMMA 16x16x4), then per-row top-128.
// - Chunked over vocab: 1024 x 8000 f32 score tile (32.8 MB) stays in 192 MB L2.
// - W (102 MB) streamed from HBM once; block's 64x128 W tile (contiguous 32 KB)
//   is staged into LDS with GLOBAL_LOAD_ASYNC_TO_LDS_B128 (ASYNCcnt path),
//   removing the 4x duplicated W reads across the 4 M-waves.
// - Steady state per K-step/wave: 2 global_load_b64 (A) + 2 ds_load_b64 (B)
//   feeding 4 v_wmma_f32_16x16x4_f32 -> loads split across VMEM + LDS pipes.
// ---------------------------------------------------------------------------

#define BATCH   1024
#define DIN     128
#define VOCAB   200000
#define TOPK    128
#define CHUNK   8000          // multiple of 64 (block N tile); 25 chunks
#define NCHUNK  (VOCAB / CHUNK)
#define CANDBUF 384           // candidate slots; arena = TOPK + CANDBUF = 512

typedef __attribute__((ext_vector_type(2))) float v2f;
typedef __attribute__((ext_vector_type(8))) float v8f;

#define WMMA_F32(a, b, c) \
    __builtin_amdgcn_wmma_f32_16x16x4_f32(false, (a), false, (b), (short)0, (c), false, false)

// --------------------------- GEMM (WMMA f32) -------------------------------
// Block: 256 threads = 8 waves laid out 4(M) x 2(N) -> 128(M) x 64(N) tile.
// Wave: 32(M) x 32(N) tile = 4 x v_wmma_f32_16x16x4_f32 per K-step, K=128.
// f32 A 16x4 layout: lane<16 holds K={0,1}, lane>=16 holds K={2,3} (v0,v1),
// i.e. a contiguous K-pair per lane -> one float2 load. B (=W^T) mirrors it.
__global__ __launch_bounds__(256) void
wmma_scores_kernel(const float* __restrict__ x, const float* __restrict__ W,
                   float* __restrict__ scores, int chunk_base) {
    __shared__ float sW[64 * DIN];          // 32 KB: block's W tile (64 rows)

    const int lane = threadIdx.x & 31;
    const int wid  = threadIdx.x >> 5;      // 0..7
    const int wm   = wid >> 1;              // 0..3  (M)
    const int wn   = wid & 1;               // 0..1  (N)
    const int m0   = blockIdx.y * 128 + wm * 32;          // batch row base
    const int nloc = blockIdx.x * 64  + wn * 32;          // chunk-local col base
    const int row  = lane & 15;
    const int half = lane >> 4;             // 0: K lo pair, 1: K hi pair

    // ---- async-stage the 64x128 W tile (contiguous 32 KB) into LDS --------
    {
        const char* gsrc = (const char*)(W + (size_t)(chunk_base + blockIdx.x * 64) * DIN);
        unsigned    lbase = (unsigned)(uintptr_t)&sW[0];
#pragma unroll
        for (int i = 0; i < 8; ++i) {       // 256 thr * 8 * 16B = 32 KB
            unsigned    off  = (threadIdx.x + 256 * i) * 16;
            unsigned    ldsa = lbase + off;
            const void* g    = gsrc + off;
            asm volatile("global_load_async_to_lds_b128 %0, %1, off"
                         :: "v"(ldsa), "v"(g) : "memory");
        }
        asm volatile("s_wait_asynccnt 0x0" ::: "memory");
    }
    __syncthreads();

    // ---- main K loop: A from global (L2-resident x), B from LDS -----------
    const float* a0p = x + (size_t)(m0 + row) * DIN + 2 * half;
    const float* a1p = a0p + 16 * DIN;
    const float* b0p = &sW[(wn * 32 + row) * DIN + 2 * half];
    const float* b1p = b0p + 16 * DIN;

    v8f c00 = {}, c01 = {}, c10 = {}, c11 = {};
#pragma unroll
    for (int k = 0; k < DIN; k += 4) {
        v2f a0 = *(const v2f*)(a0p + k);
        v2f a1 = *(const v2f*)(a1p + k);
        v2f b0 = *(const v2f*)(b0p + k);
        v2f b1 = *(const v2f*)(b1p + k);
        c00 = WMMA_F32(a0, b0, c00);
        c01 = WMMA_F32(a0, b1, c01);
        c10 = WMMA_F32(a1, b0, c10);
        c11 = WMMA_F32(a1, b1, c11);
    }

    // C layout: VGPR j -> M = j (lanes 0-15) / j+8 (lanes 16-31), N = lane%16
#pragma unroll
    for (int j = 0; j < 8; ++j) {
        const size_t r0 = (size_t)(m0 + j + 8 * half) * CHUNK;
        const size_t r1 = (size_t)(m0 + 16 + j + 8 * half) * CHUNK;
        scores[r0 + nloc + row]      = c00[j];
        scores[r0 + nloc + 16 + row] = c01[j];
        scores[r1 + nloc + row]      = c10[j];
        scores[r1 + nloc + 16 + row] = c11[j];
    }
}

// ------------------------- streaming top-k merge ---------------------------
__device__ inline void bitonic_sort_512_desc(float* v, int* id) {
    const int t = threadIdx.x;  // 256 threads, 512 elements, 1 pair/thread/stage
    for (int k = 2; k <= 512; k <<= 1) {
        for (int j = k >> 1; j > 0; j >>= 1) {
            __syncthreads();
            int i = ((t & ~(j - 1)) << 1) | (t & (j - 1));
            int l = i | j;
            bool desc = ((i & k) == 0);
            float vi = v[i], vl = v[l];
            if ((vi < vl) == desc) {
                v[i] = vl; v[l] = vi;
                int ti = id[i]; id[i] = id[l]; id[l] = ti;
            }
        }
    }
    __syncthreads();
}

__global__ __launch_bounds__(256) void
topk_merge_kernel(const float* __restrict__ scores,
                  float* __restrict__ tvals, int* __restrict__ tidx,
                  float* __restrict__ out, int chunk_base,
                  int is_first, int is_last) {
    __shared__ float sv[TOPK + CANDBUF];
    __shared__ int   si[TOPK + CANDBUF];
    __shared__ int   ccount;

    const int b = blockIdx.x;        // batch row
    const int t = threadIdx.x;       // 0..255

    // init arena: running top-128 in [0,128), candidates in [128,512)
    if (t < TOPK) {
        sv[t] = is_first ? -__builtin_inff() : tvals[(size_t)b * TOPK + t];
        si[t] = is_first ? -1 : tidx[(size_t)b * TOPK + t];
    } else {
        sv[t] = -__builtin_inff(); si[t] = -1;
    }
    sv[t + 256] = -__builtin_inff(); si[t + 256] = -1;
    if (t == 0) ccount = 0;
    __syncthreads();

    const float* rowp = scores + (size_t)b * CHUNK;
    for (int base = 0; base < CHUNK; base += TOPK) {   // 128 cols / round
        float th = sv[TOPK - 1];
        if (t < TOPK) {
            float s = rowp[base + t];
            if (s > th) {
                int p = atomicAdd(&ccount, 1);         // p <= 383 guaranteed
                sv[TOPK + p] = s;
                si[TOPK + p] = chunk_base + base + t;
            }
        }
        __syncthreads();
        if (ccount > CANDBUF - TOPK) {                 // next round might overflow
            bitonic_sort_512_desc(sv, si);
            if (t >= TOPK) { sv[t] = -__builtin_inff(); si[t] = -1; }
            sv[t + 256] = -__builtin_inff(); si[t + 256] = -1;
            if (t == 0) ccount = 0;
            __syncthreads();
        }
    }

    bitonic_sort_512_desc(sv, si);                     // final merge

    if (t < TOPK) {
        tvals[(size_t)b * TOPK + t] = sv[t];
        tidx [(size_t)b * TOPK + t] = si[t];
        if (is_last) out[(size_t)b * TOPK + t] = (float)si[t];  // exact: idx < 2^24
    }
}

// ------------------------------- launcher ----------------------------------
extern "C" void kernel_launch(void* const* d_in, const int* in_sizes, int n_in,
                              void* d_out, int out_size, void* d_ws, size_t ws_size,
                              hipStream_t stream) {
    const float* x = (const float*)d_in[0];   // [1024,128]
    const float* W = (const float*)d_in[1];   // [200000,128]
    float* out = (float*)d_out;               // [1024,128] indices (desc)

    // workspace: chunk scores | running top values | running top indices
    float* scores = (float*)d_ws;                                        // 32.8 MB
    float* tvals  = (float*)((char*)d_ws + (size_t)BATCH * CHUNK * 4);
    int*   tidx   = (int*)  ((char*)d_ws + (size_t)BATCH * CHUNK * 4
                                         + (size_t)BATCH * TOPK * 4);

    dim3 gGemm(CHUNK / 64, BATCH / 128);      // (125, 8) = 1000 blocks/chunk
    for (int c = 0; c < NCHUNK; ++c) {
        wmma_scores_kernel<<<gGemm, 256, 0, stream>>>(x, W, scores, c * CHUNK);
        topk_merge_kernel<<<BATCH, 256, 0, stream>>>(scores, tvals, tidx, out,
                                                     c * CHUNK,
                                                     c == 0, c == NCHUNK - 1);
    }
}